// SpiralDeblock_68521908241109
// MI455X (gfx1250) — compile-verified
//
#include <hip/hip_runtime.h>
#include <hip/hip_bf16.h>

typedef float v2f __attribute__((ext_vector_type(2)));
typedef float v8f __attribute__((ext_vector_type(8)));

#define B_      8
#define N_DOWN  12288
#define N_UP    49152
#define C_IN    64
#define C_OUT   32
#define SPIRAL  9
#define NNZ     147456
#define K_TOT   (SPIRAL * C_IN)   // 576

// ---------------------------------------------------------------------------
// Stage 1: sparse upsample pooling via L2 atomics.
// One thread per (nz, channel); loops over batch. x reads and atomic targets
// are contiguous across the 64 channel-threads -> fully coalesced.
// ---------------------------------------------------------------------------
__global__ void __launch_bounds__(256) pool_scatter_kernel(
    const float* __restrict__ x,
    const int*   __restrict__ up_rows,
    const int*   __restrict__ up_cols,
    const float* __restrict__ up_vals,
    float*       __restrict__ pooled) {
  int t = blockIdx.x * blockDim.x + threadIdx.x;   // NNZ*64 threads
  int c = t & (C_IN - 1);
  int i = t >> 6;
  if (i >= NNZ) return;
  int   r   = up_rows[i];
  int   col = up_cols[i];
  float v   = up_vals[i];
  const float* xp = x      + (size_t)col * C_IN + c;
  float*       pp = pooled + (size_t)r   * C_IN + c;
#pragma unroll
  for (int b = 0; b < B_; ++b) {
    atomicAdd(pp + (size_t)b * N_UP * C_IN,
              v * xp[(size_t)b * N_DOWN * C_IN]);
  }
}

// ---------------------------------------------------------------------------
// Stage 2: spiral gather + GEMM on the f32 matrix pipe.
// GEMM: M = B*N_UP rows, N = 32 (C_OUT), K = 576 (SPIRAL*C_IN).
// One wave32 -> 32 rows x 32 cols = four 16x16 f32 accumulators.
// Per V_WMMA_F32_16X16X4_F32 K-step of 4:
//   A frags (x2): lane (h = lane>>4, ln = lane&15) holds row M=ln,
//                 K = {k0+2h, k0+2h+1} -> contiguous float2 gathered from
//                 pooled[b, spiral_idx, c] (one per M-subtile)
//   B frags (x2): lane holds col N=ln(+16), K = {k0+2h, k0+2h+1}
//                 -> contiguous float2 from weight (row-major over K),
//                 SHARED by both M-subtiles (2 loads feed 4 WMMAs)
// All tiles full -> EXEC is all ones, as WMMA requires.
// ---------------------------------------------------------------------------
__global__ void __launch_bounds__(256) spiral_wmma_kernel(
    const float* __restrict__ pooled,
    const int*   __restrict__ spiral_indices,
    const float* __restrict__ weight,
    const float* __restrict__ bias,
    float*       __restrict__ out) {
  const int wave = threadIdx.x >> 5;       // 0..7
  const int lane = threadIdx.x & 31;
  const int h    = lane >> 4;              // K-pair half within wave
  const int ln   = lane & 15;              // M-row / N-col sub-index

  const long tile = (long)blockIdx.x * 8 + wave;   // 32-row tile id
  const long row0 = tile * 32;                     // flat row = b*N_UP + n
  const int  b    = (int)(row0 / N_UP);
  const int  n0   = (int)(row0 % N_UP);
  const int  nA   = n0 + ln;                       // lane's vertex, subtile 0
  const int  nB   = n0 + 16 + ln;                  // lane's vertex, subtile 1

  const float* pb = pooled + (size_t)b * N_UP * C_IN;

  v8f acc00 = {};   // rows 0..15,  cols 0..15
  v8f acc01 = {};   // rows 0..15,  cols 16..31
  v8f acc10 = {};   // rows 16..31, cols 0..15
  v8f acc11 = {};   // rows 16..31, cols 16..31

#pragma unroll 1
  for (int s = 0; s < SPIRAL; ++s) {
    const int idxA = spiral_indices[nA * SPIRAL + s];
    const int idxB = spiral_indices[nB * SPIRAL + s];
    const float* arowA = pb + (size_t)idxA * C_IN + 2 * h;
    const float* arowB = pb + (size_t)idxB * C_IN + 2 * h;
    const float* w0    = weight + (size_t)ln * K_TOT + s * C_IN + 2 * h;
    const float* w1    = w0 + (size_t)16 * K_TOT;
#pragma unroll
    for (int c0 = 0; c0 < C_IN; c0 += 4) {
      v2f a0  = *(const v2f*)(arowA + c0);
      v2f a1  = *(const v2f*)(arowB + c0);
      v2f bf0 = *(const v2f*)(w0 + c0);
      v2f bf1 = *(const v2f*)(w1 + c0);
      acc00 = __builtin_amdgcn_wmma_f32_16x16x4_f32(
                  false, a0, false, bf0, (short)0, acc00, false, false);
      acc01 = __builtin_amdgcn_wmma_f32_16x16x4_f32(
                  false, a0, false, bf1, (short)0, acc01, false, false);
      acc10 = __builtin_amdgcn_wmma_f32_16x16x4_f32(
                  false, a1, false, bf0, (short)0, acc10, false, false);
      acc11 = __builtin_amdgcn_wmma_f32_16x16x4_f32(
                  false, a1, false, bf1, (short)0, acc11, false, false);
    }
  }

  // C/D layout: VGPR v holds M = v + 8*h, N = ln (col-tile 0) / ln+16 (1).
  const float bias0 = bias[ln];
  const float bias1 = bias[ln + 16];
  float* o0 = out + (size_t)row0 * C_OUT;             // rows 0..15
  float* o1 = o0 + (size_t)16 * C_OUT;                // rows 16..31
#pragma unroll
  for (int v = 0; v < 8; ++v) {
    const int m = v + 8 * h;
    float r00 = acc00[v] + bias0; r00 = r00 > 0.f ? r00 : 0.f;
    float r01 = acc01[v] + bias1; r01 = r01 > 0.f ? r01 : 0.f;
    float r10 = acc10[v] + bias0; r10 = r10 > 0.f ? r10 : 0.f;
    float r11 = acc11[v] + bias1; r11 = r11 > 0.f ? r11 : 0.f;
    o0[(size_t)m * C_OUT + ln]      = r00;
    o0[(size_t)m * C_OUT + ln + 16] = r01;
    o1[(size_t)m * C_OUT + ln]      = r10;
    o1[(size_t)m * C_OUT + ln + 16] = r11;
  }
}

// ---------------------------------------------------------------------------
extern "C" void kernel_launch(void* const* d_in, const int* in_sizes, int n_in,
                              void* d_out, int out_size, void* d_ws, size_t ws_size,
                              hipStream_t stream) {
  const float* x              = (const float*)d_in[0];
  const int*   up_rows        = (const int*)  d_in[1];
  const int*   up_cols        = (const int*)  d_in[2];
  const float* up_vals        = (const float*)d_in[3];
  const int*   spiral_indices = (const int*)  d_in[4];
  const float* weight         = (const float*)d_in[5];
  const float* bias           = (const float*)d_in[6];
  float*       out            = (float*)d_out;

  float* pooled = (float*)d_ws;    // [B, N_UP, C_IN] f32 = ~100.7 MB scratch
  const size_t pool_bytes = (size_t)B_ * N_UP * C_IN * sizeof(float);
  hipMemsetAsync(pooled, 0, pool_bytes, stream);

  // Stage 1: NNZ*64 threads
  const int scatter_threads = NNZ * C_IN;
  pool_scatter_kernel<<<scatter_threads / 256, 256, 0, stream>>>(
      x, up_rows, up_cols, up_vals, pooled);

  // Stage 2: (B*N_UP)/32 tiles of 32 rows, 8 waves (tiles) per block
  const long n_tiles = (long)B_ * N_UP / 32;       // 12288
  const int  blocks  = (int)(n_tiles / 8);         // 1536
  spiral_wmma_kernel<<<blocks, 256, 0, stream>>>(
      pooled, spiral_indices, weight, bias, out);
}